// Model_68616397521448
// MI455X (gfx1250) — compile-verified
//
#include <hip/hip_runtime.h>

typedef __attribute__((ext_vector_type(2))) float v2f;
typedef __attribute__((ext_vector_type(8))) float v8f;

#define RULE_NUM 512
#define RES_DIM  5
#define BATCH    2048
#define FEAT     128
#define ROWS     16     // batch rows per block (one WMMA M-tile)
#define THREADS  256    // 8 wave32

__global__ __launch_bounds__(THREADS) void brb_fused_kernel(
    const float* __restrict__ x, const float* __restrict__ a,
    const float* __restrict__ b, const float* __restrict__ r,
    const float* __restrict__ dparm, float* __restrict__ out)
{
  __shared__ float s_beliefs[RULE_NUM * RES_DIM];  // softmax(b)        10240 B
  __shared__ float s_p0[RULE_NUM];                 // r - 128*E*a^2      2048 B
  __shared__ float s_p1[RULE_NUM];                 // 2*E*a              2048 B
  __shared__ float s_aw[ROWS * RULE_NUM];          // activation tile   32768 B
  __shared__ float s_red1[THREADS];
  __shared__ float s_red2[THREADS];
  __shared__ float s_s1[ROWS];
  __shared__ float s_s2[ROWS];
  __shared__ float s_sum[ROWS];
  __shared__ float s_bc[ROWS * RES_DIM];

  const int tid = threadIdx.x;
  const float E = expf(dparm[0]);

  // ---- Phase 1: per-rule constants + softmax(b) -> LDS ----
  for (int n = tid; n < RULE_NUM; n += THREADS) {
    const float av = a[n];
    s_p0[n] = r[n] - 128.0f * E * av * av;
    s_p1[n] = 2.0f * E * av;
    float bv[RES_DIM];
    float mx = -1e30f;
    #pragma unroll
    for (int k = 0; k < RES_DIM; ++k) { bv[k] = b[n * RES_DIM + k]; mx = fmaxf(mx, bv[k]); }
    float bs = 0.0f;
    #pragma unroll
    for (int k = 0; k < RES_DIM; ++k) { bv[k] = expf(bv[k] - mx); bs += bv[k]; }
    const float inv = 1.0f / bs;
    #pragma unroll
    for (int k = 0; k < RES_DIM; ++k) s_beliefs[n * RES_DIM + k] = bv[k] * inv;
  }

  // ---- Phase 2: per-batch-row S1 = sum(x/5), S2 = sum((x/5)^2) ----
  const int row = tid >> 4;        // 0..15  (batch row within tile)
  const int sub = tid & 15;        // 0..15  (16 threads cooperate per row)
  {
    const float* xr = x + (blockIdx.x * ROWS + row) * FEAT + sub * 8;
    float s1 = 0.0f, s2 = 0.0f;
    #pragma unroll
    for (int i = 0; i < 8; ++i) { const float v = xr[i] / 5.0f; s1 += v; s2 += v * v; }
    s_red1[tid] = s1;
    s_red2[tid] = s2;
  }
  __syncthreads();
  if (tid < ROWS) {
    float s1 = 0.0f, s2 = 0.0f;
    for (int i = 0; i < 16; ++i) { s1 += s_red1[tid * 16 + i]; s2 += s_red2[tid * 16 + i]; }
    s_s1[tid] = s1;
    s_s2[tid] = s2;
  }
  __syncthreads();

  // ---- Phase 3: exponent GEMM via V_WMMA_F32_16X16X4_F32, then exp() ----
  // arg[m][n] = 1*P0[n] + S1[m]*P1[n] + S2[m]*(-E) + 0
  // A (16x4, MxK): lanes 0-15 hold K={0,1} in v[0],v[1]; lanes 16-31 hold K={2,3}
  // B (4x16, KxN): lanes 0-15 hold K={0,1} for N=lane; lanes 16-31 hold K={2,3}
  // D (16x16):     vgpr j, lanes 0-15 -> M=j, N=lane; lanes 16-31 -> M=j+8, N=lane-16
  const int wave = tid >> 5;       // 0..7 -> 64 rules each
  const int lane = tid & 31;
  const int half = lane >> 4;
  const int lm   = lane & 15;

  v2f amat;
  if (half == 0) { amat.x = 1.0f;      amat.y = s_s1[lm]; }
  else           { amat.x = s_s2[lm];  amat.y = 0.0f;     }

  #pragma unroll
  for (int t = 0; t < 4; ++t) {
    const int nb = wave * 64 + t * 16;
    v2f bmat;
    if (half == 0) { bmat.x = s_p0[nb + lm]; bmat.y = s_p1[nb + lm]; }
    else           { bmat.x = -E;            bmat.y = 0.0f;          }
    v8f cm = {};
    cm = __builtin_amdgcn_wmma_f32_16x16x4_f32(
        /*neg_a=*/false, amat, /*neg_b=*/false, bmat,
        /*c_mod=*/(short)0, cm, /*reuse_a=*/false, /*reuse_b=*/false);
    #pragma unroll
    for (int j = 0; j < 8; ++j) {
      const int m = j + 8 * half;
      s_aw[m * RULE_NUM + nb + lm] = expf(cm[j]);   // aw = exp(arg)
    }
  }
  __syncthreads();

  // ---- Phase 4: s[m] = sum_n aw[m][n] ----
  {
    float p = 0.0f;
    for (int n = sub; n < RULE_NUM; n += 16) p += s_aw[row * RULE_NUM + n];
    s_red1[tid] = p;
  }
  __syncthreads();
  if (tid < ROWS) {
    float p = 0.0f;
    for (int i = 0; i < 16; ++i) p += s_red1[tid * 16 + i];
    s_sum[tid] = p;
  }
  __syncthreads();

  // ---- Phase 5: bc[m][k] = prod_n (ratio*belief + 1) - 1 ----
  float pk[RES_DIM] = {1.0f, 1.0f, 1.0f, 1.0f, 1.0f};
  {
    const float sm = s_sum[row];
    for (int n = sub; n < RULE_NUM; n += 16) {
      const float awv   = s_aw[row * RULE_NUM + n];
      const float ratio = awv / (sm - awv);
      #pragma unroll
      for (int k = 0; k < RES_DIM; ++k)
        pk[k] *= ratio * s_beliefs[n * RES_DIM + k] + 1.0f;
    }
  }
  #pragma unroll
  for (int k = 0; k < RES_DIM; ++k) {
    __syncthreads();
    s_red1[tid] = pk[k];
    __syncthreads();
    if (tid < ROWS) {
      float p = 1.0f;
      for (int i = 0; i < 16; ++i) p *= s_red1[tid * 16 + i];
      s_bc[tid * RES_DIM + k] = p - 1.0f;
    }
  }
  __syncthreads();

  // ---- Phase 6: normalize and dot with utility anchors ----
  if (tid < ROWS) {
    const float u[RES_DIM] = {-0.5f, 0.0f, 0.5f, 1.0f, 1.5f};
    float bs = 0.0f;
    #pragma unroll
    for (int k = 0; k < RES_DIM; ++k) bs += s_bc[tid * RES_DIM + k];
    float acc = 0.0f;
    #pragma unroll
    for (int k = 0; k < RES_DIM; ++k) acc += (s_bc[tid * RES_DIM + k] / bs) * u[k];
    out[blockIdx.x * ROWS + tid] = acc;
  }
}

extern "C" void kernel_launch(void* const* d_in, const int* in_sizes, int n_in,
                              void* d_out, int out_size, void* d_ws, size_t ws_size,
                              hipStream_t stream) {
  (void)in_sizes; (void)n_in; (void)out_size; (void)d_ws; (void)ws_size;
  const float* x = (const float*)d_in[0];
  const float* a = (const float*)d_in[1];
  const float* b = (const float*)d_in[2];
  const float* r = (const float*)d_in[3];
  const float* d = (const float*)d_in[4];
  float* out = (float*)d_out;
  brb_fused_kernel<<<BATCH / ROWS, THREADS, 0, stream>>>(x, a, b, r, d, out);
}